// TemporalGeneratorODE_46737834115101
// MI455X (gfx1250) — compile-verified
//
#include <hip/hip_runtime.h>

typedef float v2f __attribute__((ext_vector_type(2)));
typedef float v8f __attribute__((ext_vector_type(8)));

#define ODE_DIM   100
#define ODE_NT    7          // N padded to 112 = 7 tiles of 16
#define ODE_KI    25         // K = 100 = 25 chunks of 4
#define ODE_ROWS  64         // rows per block (4 waves x 16 rows)
#define ZSTRIDE   116        // bank-conflict-free row stride (words)
#define ODE_T     16
#define ODE_TD    (ODE_T * ODE_DIM)

__device__ __forceinline__ float ode_tanhf(float x) {
#if __has_builtin(__builtin_amdgcn_tanhf)
  return __builtin_amdgcn_tanhf(x);           // gfx1250 v_tanh_f32 (TRANS op)
#else
  float e = __expf(-2.0f * fabsf(x));
  float r = (1.0f - e) / (1.0f + e);
  return copysignf(r, x);
#endif
}

__global__ __launch_bounds__(128, 1)
void ode_rk4_wmma(const float* __restrict__ x, const float* __restrict__ W,
                  const float* __restrict__ bias, float* __restrict__ out)
{
  // W^T pre-swizzled into per-lane WMMA B fragments: [ki][nt][lane][2]
  __shared__ float wfrag[ODE_KI * ODE_NT * 64];            // 44.8 KB
  __shared__ float zbuf[ODE_ROWS * ZSTRIDE];               // 29.0 KB

  const int tid  = threadIdx.x;
  const int lane = tid & 31;
  const int wave = tid >> 5;
  const int half = lane >> 4;     // 0: lanes 0-15, 1: lanes 16-31
  const int ln   = lane & 15;

  // ---- one-time: build B fragments of W^T (B[k][n] = W[n][k]), zero-pad n>=100
  for (int idx = tid; idx < ODE_KI * ODE_NT * 32; idx += 128) {
    const int l  = idx & 31;
    const int nt = (idx >> 5) % ODE_NT;
    const int ki = idx / (32 * ODE_NT);
    const int k  = ki * 4 + ((l >> 4) << 1);   // lanes 16-31 hold K+2,K+3
    const int n  = nt * 16 + (l & 15);
    float v0 = 0.0f, v1 = 0.0f;
    if (n < ODE_DIM) { v0 = W[n * ODE_DIM + k]; v1 = W[n * ODE_DIM + k + 1]; }
    wfrag[idx * 2 + 0] = v0;
    wfrag[idx * 2 + 1] = v1;
  }
  __syncthreads();

  // bias in C-layout registers
  float breg[ODE_NT];
#pragma unroll
  for (int nt = 0; nt < ODE_NT; ++nt) {
    const int n = nt * 16 + ln;
    breg[nt] = (n < ODE_DIM) ? bias[n] : 0.0f;
  }

  const long long rowBase  = (long long)blockIdx.x * ODE_ROWS + wave * 16;
  const int       zRowBase = wave * 16;

  // RK4 state in registers, WMMA C-layout: (vgpr j, lane) -> m = j + 8*half, n = nt*16+ln
  v8f y_r[ODE_NT], sum_r[ODE_NT], t_r[ODE_NT], acc[ODE_NT];
  const v8f vzero = {0.f, 0.f, 0.f, 0.f, 0.f, 0.f, 0.f, 0.f};

  // ---- load x -> y registers + z LDS, emit frame 0
#pragma unroll
  for (int nt = 0; nt < ODE_NT; ++nt) {
    const int n = nt * 16 + ln;
#pragma unroll
    for (int j = 0; j < 8; ++j) {
      const int m = j + 8 * half;
      float v = 0.0f;
      if (n < ODE_DIM) v = x[(rowBase + m) * ODE_DIM + n];
      y_r[nt][j] = v;
      zbuf[(zRowBase + m) * ZSTRIDE + n] = v;
      if (n < ODE_DIM) out[(rowBase + m) * ODE_TD + n] = v;
    }
  }

  // A-fragment base: lane ln reads row ln, words [4*ki + 2*half, +1]
  const float* zrow = &zbuf[(zRowBase + ln) * ZSTRIDE + 2 * half];

  // GEMM: acc = z @ W^T   (175 v_wmma_f32_16x16x4_f32 per call)
  auto mm = [&]() {
    __builtin_amdgcn_wave_barrier();           // order LDS stores before frag loads
    asm volatile("" ::: "memory");
#pragma unroll
    for (int nt = 0; nt < ODE_NT; ++nt) acc[nt] = vzero;
    for (int ki = 0; ki < ODE_KI; ++ki) {
      v2f a = *(const v2f*)(zrow + 4 * ki);
#pragma unroll
      for (int nt = 0; nt < ODE_NT; ++nt) {
        v2f bf = *(const v2f*)(&wfrag[(ki * ODE_NT + nt) * 64 + lane * 2]);
        acc[nt] = __builtin_amdgcn_wmma_f32_16x16x4_f32(
            false, a, false, bf, (short)0, acc[nt], false, false);
      }
    }
    asm volatile("" ::: "memory");
  };

  const float h     = 1.0f / 15.0f;
  const float h3    = h * (1.0f / 3.0f);
  const float h8    = h * 0.125f;
  const float third = 1.0f / 3.0f;

  for (int t = 1; t < ODE_T; ++t) {
    // ---- k1 = f(y); z = y + (h/3)k1
    mm();
#pragma unroll
    for (int nt = 0; nt < ODE_NT; ++nt) {
      const int n = nt * 16 + ln;
#pragma unroll
      for (int j = 0; j < 8; ++j) {
        const int m = j + 8 * half;
        float k1 = ode_tanhf(acc[nt][j] + breg[nt]);
        sum_r[nt][j] = k1;
        t_r[nt][j]   = k1;
        zbuf[(zRowBase + m) * ZSTRIDE + n] = y_r[nt][j] + h3 * k1;
      }
    }
    // ---- k2 = f(z); z = y + h*(k2 - k1/3); t = k1 - k2; sum += 3*k2
    mm();
#pragma unroll
    for (int nt = 0; nt < ODE_NT; ++nt) {
      const int n = nt * 16 + ln;
#pragma unroll
      for (int j = 0; j < 8; ++j) {
        const int m = j + 8 * half;
        float k2 = ode_tanhf(acc[nt][j] + breg[nt]);
        float zv = y_r[nt][j] + h * (k2 - t_r[nt][j] * third);
        t_r[nt][j]   = t_r[nt][j] - k2;
        sum_r[nt][j] = sum_r[nt][j] + 3.0f * k2;
        zbuf[(zRowBase + m) * ZSTRIDE + n] = zv;
      }
    }
    // ---- k3 = f(z); z = y + h*(k1 - k2 + k3); sum += 3*k3
    mm();
#pragma unroll
    for (int nt = 0; nt < ODE_NT; ++nt) {
      const int n = nt * 16 + ln;
#pragma unroll
      for (int j = 0; j < 8; ++j) {
        const int m = j + 8 * half;
        float k3 = ode_tanhf(acc[nt][j] + breg[nt]);
        sum_r[nt][j] = sum_r[nt][j] + 3.0f * k3;
        zbuf[(zRowBase + m) * ZSTRIDE + n] = y_r[nt][j] + h * (t_r[nt][j] + k3);
      }
    }
    // ---- k4 = f(z); y += (sum + k4)*h/8; emit frame t; z = y for next step
    mm();
#pragma unroll
    for (int nt = 0; nt < ODE_NT; ++nt) {
      const int n = nt * 16 + ln;
#pragma unroll
      for (int j = 0; j < 8; ++j) {
        const int m = j + 8 * half;
        float k4 = ode_tanhf(acc[nt][j] + breg[nt]);
        float yn = y_r[nt][j] + (sum_r[nt][j] + k4) * h8;
        y_r[nt][j] = yn;
        zbuf[(zRowBase + m) * ZSTRIDE + n] = yn;
        if (n < ODE_DIM) out[(rowBase + m) * ODE_TD + (long long)t * ODE_DIM + n] = yn;
      }
    }
  }
}

extern "C" void kernel_launch(void* const* d_in, const int* in_sizes, int n_in,
                              void* d_out, int out_size, void* d_ws, size_t ws_size,
                              hipStream_t stream) {
  (void)n_in; (void)out_size; (void)d_ws; (void)ws_size;
  const float* x = (const float*)d_in[0];   // [65536, 100]
  const float* W = (const float*)d_in[1];   // [100, 100]
  const float* b = (const float*)d_in[2];   // [100]
  float* out = (float*)d_out;               // [65536, 16, 100]
  const int batch = in_sizes[0] / ODE_DIM;
  const int grid  = (batch + ODE_ROWS - 1) / ODE_ROWS;
  ode_rk4_wmma<<<grid, 128, 0, stream>>>(x, W, b, out);
}